// BlurredBoundaryAdaptor_52819507806776
// MI455X (gfx1250) — compile-verified
//
#include <hip/hip_runtime.h>
#include <math.h>

// Problem constants (from reference setup_inputs)
constexpr int B_ = 8, T_ = 8192, D_ = 512, N_ = 512;
constexpr float DROP_P_ = 0.8f;
constexpr int TT   = 16;        // time-tile per block
constexpr int ROWS = TT + 2;    // +1 halo each side
constexpr int DQ   = D_ / 4;    // float4 groups per row = 128

typedef int v4i __attribute__((ext_vector_type(4)));
typedef __attribute__((address_space(3))) void lds_void_t;
typedef __attribute__((address_space(1))) void glb_void_t;
typedef __attribute__((address_space(1))) v4i  glb_v4i_t;
typedef __attribute__((address_space(3))) v4i  lds_v4i_t;

// ---------------------------------------------------------------------------
// Async global -> LDS b128 copy (CDNA5 DMA path), with inline-asm fallback.
// ---------------------------------------------------------------------------
__device__ __forceinline__ void async_copy_b128(const float* gsrc, float* lds_dst) {
#if __has_builtin(__builtin_amdgcn_global_load_async_to_lds_b128)
    __builtin_amdgcn_global_load_async_to_lds_b128(
        (glb_v4i_t*)(glb_void_t*)gsrc,
        (lds_v4i_t*)(lds_void_t*)lds_dst,
        /*offset=*/0, /*cpol=*/0);
#else
    unsigned lds_addr = (unsigned)(unsigned long long)(lds_void_t*)lds_dst;
    asm volatile("global_load_async_to_lds_b128 %0, %1, off"
                 :: "v"(lds_addr), "v"(gsrc) : "memory");
#endif
}

__device__ __forceinline__ void async_wait_all() {
#if __has_builtin(__builtin_amdgcn_s_wait_asynccnt)
    __builtin_amdgcn_s_wait_asynccnt(0);
#else
    asm volatile("s_wait_asynccnt 0" ::: "memory");
#endif
}

__device__ __forceinline__ float silu_f(float x) {
    return x / (1.0f + __expf(-x));
}

// ---------------------------------------------------------------------------
// Kernel A (defined first so its asm leads the disasm snippet):
// fused mask * conv1d_depthwise(k=3, SAME) + bias -> silu -> residual.
// One block = one (b, 16-timestep) tile. 256 threads (8 waves).
// Async DMA for the 18x512 f32 tile is issued FIRST so the transfer overlaps
// the synchronous weight/bias/mask staging that follows.
// ---------------------------------------------------------------------------
__global__ __launch_bounds__(256) void fused_kernel(const float* __restrict__ x,
                                                    const float* __restrict__ w,    // [D,1,3]
                                                    const float* __restrict__ bias, // [D]
                                                    const float* __restrict__ mask, // [B,T]
                                                    float* __restrict__ out) {
    __shared__ __align__(16) float lds_c[ROWS * D_];   // raw (unmasked) tile rows
    __shared__ __align__(16) float lds_w[3 * D_];      // transposed weights [k][d]
    __shared__ __align__(16) float lds_bias[D_];
    __shared__ float lds_m[ROWS];

    const int tid  = threadIdx.x;
    const int tile = blockIdx.x;                 // B * (T/TT) tiles
    const int b    = tile / (T_ / TT);
    const int t0   = (tile % (T_ / TT)) * TT;
    const size_t base = (size_t)b * T_ * D_;

    // 1) issue async staging of 18 time rows x 512 ch (b128 granularity).
    //    Halo rows use clamped addresses; their mask is forced to 0 below,
    //    which reproduces SAME zero-padding exactly.
    #pragma unroll
    for (int i = tid; i < ROWS * DQ; i += 256) {   // 2304 copies, 9 per thread
        const int r  = i >> 7;                     // row 0..17
        const int dq = i & (DQ - 1);               // float4 group
        int gt = t0 - 1 + r;
        gt = gt < 0 ? 0 : (gt >= T_ ? T_ - 1 : gt);
        async_copy_b128(x + base + (size_t)gt * D_ + dq * 4,
                        &lds_c[r * D_ + dq * 4]);
    }

    // 2) overlap: stage weights (transpose [D][3] -> [3][D]), bias, mask row
    #pragma unroll
    for (int k = 0; k < 3; ++k)
        for (int d = tid; d < D_; d += 256)
            lds_w[k * D_ + d] = w[d * 3 + k];
    for (int d = tid; d < D_; d += 256) lds_bias[d] = bias[d];
    if (tid < ROWS) {
        const int gt = t0 - 1 + tid;
        lds_m[tid] = (gt >= 0 && gt < T_) ? mask[b * T_ + gt] : 0.0f;
    }

    // 3) wait for DMA + make LDS visible to all waves
    async_wait_all();
    __syncthreads();

    // 4) compute: each thread owns a fixed float4 channel group
    const int dq = tid & (DQ - 1);
    const int d4 = dq * 4;
    const float4 w0 = *(const float4*)&lds_w[0 * D_ + d4];
    const float4 w1 = *(const float4*)&lds_w[1 * D_ + d4];
    const float4 w2 = *(const float4*)&lds_w[2 * D_ + d4];
    const float4 bb = *(const float4*)&lds_bias[d4];

    for (int j = (tid >> 7); j < TT; j += 2) {     // 8 iters, uniform, full EXEC
        const float m0 = lds_m[j], m1 = lds_m[j + 1], m2 = lds_m[j + 2];
        float4 a = *(const float4*)&lds_c[(j    ) * D_ + d4];
        float4 c = *(const float4*)&lds_c[(j + 1) * D_ + d4];
        float4 e = *(const float4*)&lds_c[(j + 2) * D_ + d4];
        a.x *= m0; a.y *= m0; a.z *= m0; a.w *= m0;
        c.x *= m1; c.y *= m1; c.z *= m1; c.w *= m1;
        e.x *= m2; e.y *= m2; e.z *= m2; e.w *= m2;

        float4 y;
        y.x = fmaf(w0.x, a.x, fmaf(w1.x, c.x, fmaf(w2.x, e.x, bb.x)));
        y.y = fmaf(w0.y, a.y, fmaf(w1.y, c.y, fmaf(w2.y, e.y, bb.y)));
        y.z = fmaf(w0.z, a.z, fmaf(w1.z, c.z, fmaf(w2.z, e.z, bb.z)));
        y.w = fmaf(w0.w, a.w, fmaf(w1.w, c.w, fmaf(w2.w, e.w, bb.w)));

        float4 o;
        o.x = c.x + silu_f(y.x);
        o.y = c.y + silu_f(y.y);
        o.z = c.z + silu_f(y.z);
        o.w = c.w + silu_f(y.w);

        *(float4*)&out[base + (size_t)(t0 + j) * D_ + d4] = o;
    }
}

// ---------------------------------------------------------------------------
// Kernel B: boundary mask [B, T] in workspace.
// mask = 1 everywhere; mask[b, cumsum(dur)[b,n]] = 0 where drop_rand < 0.8
// ---------------------------------------------------------------------------
__global__ __launch_bounds__(512) void mask_kernel(const int* __restrict__ dur,
                                                   const float* __restrict__ drop_rand,
                                                   float* __restrict__ mask) {
    const int b = blockIdx.x;
    const int n = threadIdx.x;            // 512 == N_
    __shared__ int s[N_];

    // init mask row to 1.0
    for (int t = n; t < T_; t += 512) mask[b * T_ + t] = 1.0f;

    s[n] = dur[b * N_ + n];
    __threadfence_block();
    __syncthreads();

    // Hillis-Steele inclusive scan over N=512
    #pragma unroll
    for (int off = 1; off < N_; off <<= 1) {
        int v = (n >= off) ? s[n - off] : 0;
        __syncthreads();
        s[n] += v;
        __syncthreads();
    }

    const int boundary = s[n];            // inclusive cumsum
    const float r = drop_rand[b * N_ + n];
    if (r < DROP_P_ && boundary >= 0 && boundary < T_) {
        mask[b * T_ + boundary] = 0.0f;   // duplicates all write 0: benign
    }
}

// ---------------------------------------------------------------------------
extern "C" void kernel_launch(void* const* d_in, const int* in_sizes, int n_in,
                              void* d_out, int out_size, void* d_ws, size_t ws_size,
                              hipStream_t stream) {
    (void)in_sizes; (void)n_in; (void)out_size; (void)ws_size;
    const float* c_text    = (const float*)d_in[0];   // [B,T,D] f32
    const int*   dur       = (const int*)  d_in[1];   // [B,N] i32
    const float* drop_rand = (const float*)d_in[2];   // [B,N] f32
    const float* blur_w    = (const float*)d_in[3];   // [D,1,3] f32
    const float* blur_b    = (const float*)d_in[4];   // [D] f32
    float* out  = (float*)d_out;
    float* mask = (float*)d_ws;                       // B*T floats = 256 KB scratch

    mask_kernel<<<B_, 512, 0, stream>>>(dur, drop_rand, mask);
    fused_kernel<<<B_ * (T_ / TT), 256, 0, stream>>>(c_text, blur_w, blur_b, mask, out);
}